// DAGNN_5600637354060
// MI455X (gfx1250) — compile-verified
//
#include <hip/hip_runtime.h>
#include <math.h>

#define F_DIM 128
#define H_DIM 256
#define C_DIM 40
#define K_HOPS 10

typedef __attribute__((ext_vector_type(2))) float v2f;
typedef __attribute__((ext_vector_type(8))) float v8f;

// ---------------- degree / norm ----------------

__global__ void deg_init_kernel(float* __restrict__ deg, int n) {
    int i = blockIdx.x * blockDim.x + threadIdx.x;
    if (i < n) deg[i] = 1.0f;   // self-loop contributes 1
}

__global__ void deg_edge_kernel(const long long* __restrict__ ei, float* __restrict__ deg,
                                long long E) {
    long long e = (long long)blockIdx.x * blockDim.x + threadIdx.x;
    if (e < E) {
        int col = (int)ei[E + e];
        atomicAdd(&deg[col], 1.0f);
    }
}

__global__ void deg_fin_kernel(float* __restrict__ dinv, int n) {
    int i = blockIdx.x * blockDim.x + threadIdx.x;
    if (i < n) {
        float d = dinv[i];
        dinv[i] = (d > 0.0f) ? rsqrtf(d) : 0.0f;
    }
}

__global__ void norm_kernel(const long long* __restrict__ ei, const float* __restrict__ dinv,
                            float* __restrict__ nrm, long long E) {
    long long e = (long long)blockIdx.x * blockDim.x + threadIdx.x;
    if (e < E) {
        int row = (int)ei[e];
        int col = (int)ei[E + e];
        nrm[e] = dinv[row] * dinv[col];
    }
}

// ---------------- fused MLP with f32 WMMA ----------------
// block = 256 threads (8 waves), tile = 16 rows of x.
// GEMM1: each wave owns two 16-col tiles of H=256; K=128 in 16x16x4 steps.
// GEMM2: waves 0..2 cover cols 0..47 (valid 0..39); K=256.

#define XS 132   // padded LDS stride for x tile (128%64==0 would 16-way conflict)
#define HS 260   // padded LDS stride for h1 tile

__global__ __launch_bounds__(256) void mlp_wmma_kernel(
        const float* __restrict__ x,
        const float* __restrict__ W1, const float* __restrict__ b1,
        const float* __restrict__ W2, const float* __restrict__ b2,
        float* __restrict__ h0, int n) {
    __shared__ float sx[16 * XS];
    __shared__ float sh[16 * HS];

    const int tid  = threadIdx.x;
    const int wave = tid >> 5;
    const int lane = tid & 31;
    const int rowBase = blockIdx.x * 16;

    // cooperative load of the 16x128 x tile
    for (int i = tid; i < 16 * F_DIM; i += 256) {
        int r = i / F_DIM, c = i % F_DIM;
        int gr = rowBase + r;
        sx[r * XS + c] = (gr < n) ? x[(size_t)gr * F_DIM + c] : 0.0f;
    }
    __syncthreads();

    const int mrow  = lane & 15;         // M (A) / N (B) index within 16
    const int khalf = (lane >> 4) * 2;   // lanes 0-15 hold K=k,k+1 ; 16-31 hold K=k+2,k+3
    const int rofs  = (lane >> 4) * 8;   // C/D rows: VGPR j -> M = j (+8 for hi lanes)

    // ---- GEMM1 ----
    v8f acc0 = {};
    v8f acc1 = {};
    const int col0 = wave * 16;
    const int col1 = (wave + 8) * 16;
    for (int k = 0; k < F_DIM; k += 4) {
        int kk = k + khalf;
        v2f a;
        a.x = sx[mrow * XS + kk];
        a.y = sx[mrow * XS + kk + 1];
        v2f bA;
        bA.x = W1[(size_t)kk * H_DIM + col0 + mrow];
        bA.y = W1[(size_t)(kk + 1) * H_DIM + col0 + mrow];
        v2f bB;
        bB.x = W1[(size_t)kk * H_DIM + col1 + mrow];
        bB.y = W1[(size_t)(kk + 1) * H_DIM + col1 + mrow];
        acc0 = __builtin_amdgcn_wmma_f32_16x16x4_f32(false, a, false, bA, (short)0, acc0, false, false);
        acc1 = __builtin_amdgcn_wmma_f32_16x16x4_f32(false, a, false, bB, (short)0, acc1, false, false);
    }

    // bias + relu, stage h1 in LDS
    {
        float bia0 = b1[col0 + mrow];
        float bia1 = b1[col1 + mrow];
        for (int j = 0; j < 8; ++j) {
            int r = j + rofs;
            sh[r * HS + col0 + mrow] = fmaxf(acc0[j] + bia0, 0.0f);
            sh[r * HS + col1 + mrow] = fmaxf(acc1[j] + bia1, 0.0f);
        }
    }
    __syncthreads();

    // ---- GEMM2 ----
    if (wave < 3) {
        v8f acc = {};
        const int ct = wave * 16;
        const int c  = ct + mrow;
        const bool cv = (c < C_DIM);
        for (int k = 0; k < H_DIM; k += 4) {
            int kk = k + khalf;
            v2f a;
            a.x = sh[mrow * HS + kk];
            a.y = sh[mrow * HS + kk + 1];
            v2f b;
            b.x = cv ? W2[(size_t)kk * C_DIM + c] : 0.0f;
            b.y = cv ? W2[(size_t)(kk + 1) * C_DIM + c] : 0.0f;
            acc = __builtin_amdgcn_wmma_f32_16x16x4_f32(false, a, false, b, (short)0, acc, false, false);
        }
        if (cv) {
            float bias = b2[c];
            for (int j = 0; j < 8; ++j) {
                int gr = rowBase + j + rofs;
                if (gr < n) h0[(size_t)gr * C_DIM + c] = acc[j] + bias;
            }
        }
    }
}

// ---------------- propagation ----------------

// dst = dinv^2 * src  (self-loop term; also initializes the plane)
__global__ void selfloop_kernel(const float* __restrict__ src, float* __restrict__ dst,
                                const float* __restrict__ dinv, int n) {
    int idx = blockIdx.x * blockDim.x + threadIdx.x;
    if (idx < n * C_DIM) {
        int node = idx / C_DIM;
        float w = dinv[node];
        dst[idx] = w * w * src[idx];
    }
}

// 8 lanes per edge, 5 channels per lane; L2-resident f32 atomics
__global__ void edge_prop_kernel(const long long* __restrict__ ei,
                                 const float* __restrict__ nrm,
                                 const float* __restrict__ src,
                                 float* __restrict__ dst, long long E) {
    long long idx = (long long)blockIdx.x * blockDim.x + threadIdx.x;
    long long e = idx >> 3;
    int r = (int)(idx & 7);
    if (e < E) {
        int row = (int)ei[e];
        int col = (int)ei[E + e];
        float w = nrm[e];
        const float* s = src + (size_t)row * C_DIM;
        float* d = dst + (size_t)col * C_DIM;
        #pragma unroll
        for (int j = 0; j < 5; ++j) {
            int c = r + 8 * j;
            atomicAdd(&d[c], w * s[c]);
        }
    }
}

// ---------------- attention combine + log_softmax ----------------

__device__ __forceinline__ float wave_sum(float v) {
    for (int m = 16; m >= 1; m >>= 1) v += __shfl_xor(v, m, 32);
    return v;
}
__device__ __forceinline__ float wave_max(float v) {
    for (int m = 16; m >= 1; m >>= 1) v = fmaxf(v, __shfl_xor(v, m, 32));
    return v;
}

// one wave per node: lane handles channels lane and lane+32 (lane<8)
__global__ __launch_bounds__(256) void combine_kernel(
        const float* __restrict__ pps,            // [K+1][N][C]
        const float* __restrict__ proj_w,         // [C]
        const float* __restrict__ proj_b,         // [1]
        float* __restrict__ out, int n) {
    int node = blockIdx.x * 8 + (threadIdx.x >> 5);
    int lane = threadIdx.x & 31;
    if (node >= n) return;

    const int   c0 = lane;
    const int   c1 = 32 + lane;
    const bool  hi = (lane < 8);
    const float pw0 = proj_w[c0];
    const float pw1 = hi ? proj_w[c1] : 0.0f;
    const float pb  = proj_b[0];

    float acc0 = 0.0f, acc1 = 0.0f;
    const size_t plane = (size_t)n * C_DIM;
    const float* base = pps + (size_t)node * C_DIM;
    for (int k = 0; k <= K_HOPS; ++k) {
        float p0 = base[c0];
        float p1 = hi ? base[c1] : 0.0f;
        float s  = wave_sum(p0 * pw0 + p1 * pw1);
        float score = 1.0f / (1.0f + expf(-(s + pb)));
        acc0 += score * p0;
        acc1 += score * p1;
        base += plane;
    }

    // log_softmax over 40 valid channels
    float m  = wave_max(hi ? fmaxf(acc0, acc1) : acc0);
    float se = wave_sum(expf(acc0 - m) + (hi ? expf(acc1 - m) : 0.0f));
    float ls = logf(se);
    out[(size_t)node * C_DIM + c0] = acc0 - m - ls;
    if (hi) out[(size_t)node * C_DIM + c1] = acc1 - m - ls;
}

// ---------------- launcher ----------------

extern "C" void kernel_launch(void* const* d_in, const int* in_sizes, int n_in,
                              void* d_out, int out_size, void* d_ws, size_t ws_size,
                              hipStream_t stream) {
    const float*     x      = (const float*)d_in[0];
    const long long* ei     = (const long long*)d_in[1];
    const float*     W1     = (const float*)d_in[2];
    const float*     b1     = (const float*)d_in[3];
    const float*     W2     = (const float*)d_in[4];
    const float*     b2     = (const float*)d_in[5];
    const float*     proj_w = (const float*)d_in[6];
    const float*     proj_b = (const float*)d_in[7];
    float*           out    = (float*)d_out;

    const int       n = in_sizes[0] / F_DIM;     // 100000
    const long long E = in_sizes[1] / 2;         // 3200000

    // workspace: dinv[n] | norm[E] | pps[(K+1) * n * C]
    float* dinv = (float*)d_ws;
    float* nrm  = dinv + n;
    float* pps  = nrm + E;
    const size_t plane = (size_t)n * C_DIM;

    // degrees -> d^-1/2 -> per-edge norm
    deg_init_kernel<<<(n + 255) / 256, 256, 0, stream>>>(dinv, n);
    deg_edge_kernel<<<(unsigned)((E + 255) / 256), 256, 0, stream>>>(ei, dinv, E);
    deg_fin_kernel<<<(n + 255) / 256, 256, 0, stream>>>(dinv, n);
    norm_kernel<<<(unsigned)((E + 255) / 256), 256, 0, stream>>>(ei, dinv, nrm, E);

    // fused MLP -> pps plane 0
    mlp_wmma_kernel<<<(n + 15) / 16, 256, 0, stream>>>(x, W1, b1, W2, b2, pps, n);

    // K hops of normalized propagation
    const unsigned sl_blocks = (unsigned)(((size_t)n * C_DIM + 255) / 256);
    const unsigned ep_blocks = (unsigned)((E * 8 + 255) / 256);
    for (int k = 1; k <= K_HOPS; ++k) {
        const float* src = pps + (size_t)(k - 1) * plane;
        float*       dst = pps + (size_t)k * plane;
        selfloop_kernel<<<sl_blocks, 256, 0, stream>>>(src, dst, dinv, n);
        edge_prop_kernel<<<ep_blocks, 256, 0, stream>>>(ei, nrm, src, dst, E);
    }

    // attention-gated combine + log_softmax
    combine_kernel<<<(n + 7) / 8, 256, 0, stream>>>(pps, proj_w, proj_b, out, n);
}